// Policy_3530463117663
// MI455X (gfx1250) — compile-verified
//
#include <hip/hip_runtime.h>

#define N_NODES   10000
#define N_EDGES   160000
#define FDIM      512
#define NEG_SLOPE 0.2f

typedef __attribute__((ext_vector_type(2))) float v2f;
typedef __attribute__((ext_vector_type(8))) float v8f;

// ---------------------------------------------------------------------------
// Kernel 0: init accumulators (out_agg=0, denom=0, amax=-inf bits)
// ---------------------------------------------------------------------------
__global__ void gat_init(float* __restrict__ out_agg,
                         float* __restrict__ denom,
                         unsigned int* __restrict__ amax_bits) {
  const int stride = gridDim.x * blockDim.x;
  for (int i = blockIdx.x * blockDim.x + threadIdx.x; i < N_NODES * FDIM; i += stride)
    out_agg[i] = 0.0f;
  for (int i = blockIdx.x * blockDim.x + threadIdx.x; i < N_NODES; i += stride) {
    denom[i] = 0.0f;
    amax_bits[i] = 0xFF800000u;   // -inf
  }
}

// ---------------------------------------------------------------------------
// Kernel 1: project attention vectors through W once:
//   w_s[k] = sum_f W[k,f] * att_src[f]   (k = 0..3), likewise w_d.
// Algebra: a_src = (x@W)@att_src = x@(W@att_src) -> per-node logits become a
// 4-element dot, eliminating any re-read of h and all cross-lane reductions
// in the hot path. 8 waves: wave w handles (k = w&3, src/dst = w>>2).
// ---------------------------------------------------------------------------
__global__ void gat_att_proj(const float* __restrict__ W,
                             const float* __restrict__ att_src,
                             const float* __restrict__ att_dst,
                             float* __restrict__ w_s,   // [4]
                             float* __restrict__ w_d) { // [4]
  const int lane = threadIdx.x & 31;
  const int wave = threadIdx.x >> 5;      // 0..7
  const int k    = wave & 3;
  const bool is_dst = (wave >> 2) != 0;
  const float* att = is_dst ? att_dst : att_src;
  const float* Wr  = W + k * FDIM;
  float p = 0.0f;
#pragma unroll 4
  for (int f = lane; f < FDIM; f += 32)
    p += Wr[f] * att[f];
#pragma unroll
  for (int off = 16; off > 0; off >>= 1)
    p += __shfl_xor(p, off, 32);
  if (lane == 0) {
    if (is_dst) w_d[k] = p; else w_s[k] = p;
  }
}

// ---------------------------------------------------------------------------
// Kernel 2: h = x @ W via V_WMMA_F32_16X16X4_F32 (K=4 == x's 4 cols).
// 625 row-tiles x 32 col-tiles = 20000 tiles. Each wave owns one col tile
// (B = 16 cols of W, loaded once) and 5 row tiles -> 5 unrolled WMMAs.
// 125 row-groups * 32 col-tiles = 4000 waves, 4 waves/block -> 1000 blocks.
// ---------------------------------------------------------------------------
__global__ void gat_xw_wmma(const float* __restrict__ x,
                            const float* __restrict__ W,
                            float* __restrict__ h) {
  const int lane = threadIdx.x & 31;
  const int wave = threadIdx.x >> 5;
  const int wg   = blockIdx.x * 4 + wave;   // 0..3999
  const int tcol   = wg & 31;               // 0..31  (16 features per tile)
  const int rgroup = wg >> 5;               // 0..124 (5 row tiles each)

  const int nloc = lane & 15;
  const int col  = tcol * 16 + nloc;        // feature column this lane owns
  const int k0   = (lane >= 16) ? 2 : 0;    // K split across lane halves

  // B: 4x16 f32 tile of W, VGPR0 = row K=k0, VGPR1 = row K=k0+1
  v2f b;
  b.x = W[(k0    ) * FDIM + col];
  b.y = W[(k0 + 1) * FDIM + col];

  const int half_hi = (lane >= 16) ? 8 : 0;

#pragma unroll
  for (int i = 0; i < 5; ++i) {
    const int trow = rgroup * 5 + i;        // 0..624
    const int m    = trow * 16 + nloc;      // node row this lane supplies for A

    // A: 16x4 f32, lane holds (M=m, K=k0) and (M=m, K=k0+1)
    v2f a;
    a.x = x[m * 4 + k0];
    a.y = x[m * 4 + k0 + 1];

    v8f c = {};
    c = __builtin_amdgcn_wmma_f32_16x16x4_f32(
        /*neg_a=*/false, a, /*neg_b=*/false, b,
        /*c_mod=*/(short)0, c, /*reuse_a=*/false, /*reuse_b=*/false);

    // D layout: VGPR r -> M=r (lanes 0-15) / M=8+r (lanes 16-31), N=lane%16
    const int row0 = trow * 16 + half_hi;
#pragma unroll
    for (int r = 0; r < 8; ++r)
      h[(size_t)(row0 + r) * FDIM + col] = c[r];
  }
}

// ---------------------------------------------------------------------------
// Kernel 3: per-node logits from the projected attention vectors:
//   a_src[n] = dot(x[n,0:4], w_s), a_dst[n] = dot(x[n,0:4], w_d).
// One thread per node; x row is a single b128 load.
// ---------------------------------------------------------------------------
__global__ void gat_node_logits(const float* __restrict__ x,
                                const float* __restrict__ w_s,
                                const float* __restrict__ w_d,
                                float* __restrict__ a_src,
                                float* __restrict__ a_dst) {
  const int n = blockIdx.x * blockDim.x + threadIdx.x;
  if (n >= N_NODES) return;
  const float4 xv = ((const float4*)x)[n];
  a_src[n] = xv.x * w_s[0] + xv.y * w_s[1] + xv.z * w_s[2] + xv.w * w_s[3];
  a_dst[n] = xv.x * w_d[0] + xv.y * w_d[1] + xv.z * w_d[2] + xv.w * w_d[3];
}

// float atomic-max via sign-split integer trick (valid incl. -inf init)
__device__ __forceinline__ void atomicMaxF32(float* addr, float val) {
  if (val >= 0.0f)
    atomicMax((int*)addr, __float_as_int(val));
  else
    atomicMin((unsigned int*)addr, (unsigned int)__float_as_int(val));
}

// ---------------------------------------------------------------------------
// Kernel 4: per-edge raw logits + segment max over dst
// ---------------------------------------------------------------------------
__global__ void gat_edge_alpha(const int* __restrict__ src,
                               const int* __restrict__ dst,
                               const float* __restrict__ a_src,
                               const float* __restrict__ a_dst,
                               const float* __restrict__ noise,
                               float* __restrict__ alpha,
                               float* __restrict__ amax) {
  const int e = blockIdx.x * blockDim.x + threadIdx.x;
  if (e >= N_EDGES) return;
  const int s = src[e], d = dst[e];
  float a = a_src[s] + a_dst[d];
  a = (a > 0.0f) ? a : NEG_SLOPE * a;   // leaky relu
  a += noise[e];
  alpha[e] = a;
  atomicMaxF32(&amax[d], a);
}

// ---------------------------------------------------------------------------
// Kernel 5: e = exp(alpha - amax[dst]) (amax -inf -> 0), segment-sum denom
// ---------------------------------------------------------------------------
__global__ void gat_edge_exp(const int* __restrict__ dst,
                             float* __restrict__ alpha,   // in: logits, out: exp
                             const float* __restrict__ amax,
                             float* __restrict__ denom) {
  const int e = blockIdx.x * blockDim.x + threadIdx.x;
  if (e >= N_EDGES) return;
  const int d = dst[e];
  float am = amax[d];
  am = (am < -3.0e38f) ? 0.0f : am;     // where(isfinite(amax), amax, 0)
  const float ev = __expf(alpha[e] - am);
  alpha[e] = ev;
  atomicAdd(&denom[d], ev);
}

// ---------------------------------------------------------------------------
// Kernel 6: out_agg[dst,:] += (e/denom[dst]) * h[src,:]
// One 128-thread block per edge; B128 gather from L2-resident h,
// 4 f32 atomics per thread into L2-resident out_agg. Dominant phase.
// ---------------------------------------------------------------------------
__global__ void gat_aggregate(const int* __restrict__ src,
                              const int* __restrict__ dst,
                              const float* __restrict__ eexp,
                              const float* __restrict__ denom,
                              const float* __restrict__ h,
                              float* __restrict__ out_agg) {
  const int edge = blockIdx.x;
  const int t = threadIdx.x;                        // 0..127 -> 4 floats each
  const int s = src[edge], d = dst[edge];
  const float coef = eexp[edge] / (denom[d] + 1e-16f);
  const float4 v = ((const float4*)(h + (size_t)s * FDIM))[t];
  float* od = out_agg + (size_t)d * FDIM + t * 4;
  atomicAdd(od + 0, coef * v.x);
  atomicAdd(od + 1, coef * v.y);
  atomicAdd(od + 2, coef * v.z);
  atomicAdd(od + 3, coef * v.w);
}

// ---------------------------------------------------------------------------
// Kernel 7: y = tanh(out_agg @ W_out), W_out is (512,2). One wave per node.
// ---------------------------------------------------------------------------
__global__ void gat_out_proj(const float* __restrict__ out_agg,
                             const float* __restrict__ W_out,
                             float* __restrict__ y) {
  const int lane = threadIdx.x & 31;
  const int node = (blockIdx.x * blockDim.x + threadIdx.x) >> 5;
  if (node >= N_NODES) return;
  const float* o = out_agg + (size_t)node * FDIM;
  float p0 = 0.0f, p1 = 0.0f;
#pragma unroll 4
  for (int f = lane; f < FDIM; f += 32) {
    const float v = o[f];
    p0 += v * W_out[f * 2 + 0];
    p1 += v * W_out[f * 2 + 1];
  }
#pragma unroll
  for (int off = 16; off > 0; off >>= 1) {
    p0 += __shfl_xor(p0, off, 32);
    p1 += __shfl_xor(p1, off, 32);
  }
  if (lane == 0) {
    y[node * 2 + 0] = tanhf(p0);
    y[node * 2 + 1] = tanhf(p1);
  }
}

// ---------------------------------------------------------------------------
extern "C" void kernel_launch(void* const* d_in, const int* in_sizes, int n_in,
                              void* d_out, int out_size, void* d_ws, size_t ws_size,
                              hipStream_t stream) {
  (void)in_sizes; (void)n_in; (void)out_size; (void)ws_size;

  const float* x          = (const float*)d_in[0];          // (N,4)
  const int*   edge_index = (const int*)  d_in[1];          // (2,E)
  const float* noise      = (const float*)d_in[2];          // (E,1)
  const float* W          = (const float*)d_in[3];          // (4,512)
  const float* att_src    = (const float*)d_in[4];          // (1,512)
  const float* att_dst    = (const float*)d_in[5];          // (1,512)
  const float* W_out      = (const float*)d_in[6];          // (512,2)
  float*       y          = (float*)d_out;                  // (N,2)

  const int* src = edge_index;             // row 0
  const int* dst = edge_index + N_EDGES;   // row 1

  // workspace carve-up (floats); total ~41.7 MB
  float* ws      = (float*)d_ws;
  float* h       = ws;                                   // N*F
  float* out_agg = h       + (size_t)N_NODES * FDIM;     // N*F
  float* a_src   = out_agg + (size_t)N_NODES * FDIM;     // N
  float* a_dst   = a_src   + N_NODES;                    // N
  float* amax    = a_dst   + N_NODES;                    // N
  float* denom   = amax    + N_NODES;                    // N
  float* alpha   = denom   + N_NODES;                    // E
  float* w_s     = alpha   + N_EDGES;                    // 4
  float* w_d     = w_s     + 4;                          // 4

  // 0: init accumulators
  gat_init<<<2048, 256, 0, stream>>>(out_agg, denom, (unsigned int*)amax);

  // 1: project attention vectors through W (w_s = W@att_src, w_d = W@att_dst)
  gat_att_proj<<<1, 256, 0, stream>>>(W, att_src, att_dst, w_s, w_d);

  // 2: h = x @ W via WMMA f32 16x16x4 (4000 waves, 5 tiles each)
  gat_xw_wmma<<<1000, 128, 0, stream>>>(x, W, h);

  // 3: per-node logits from 4-element dots (a_src = x@w_s, a_dst = x@w_d)
  gat_node_logits<<<(N_NODES + 255) / 256, 256, 0, stream>>>(x, w_s, w_d,
                                                             a_src, a_dst);

  // 4: edge logits + segment max
  gat_edge_alpha<<<(N_EDGES + 255) / 256, 256, 0, stream>>>(
      src, dst, a_src, a_dst, noise, alpha, amax);

  // 5: exp + segment sum
  gat_edge_exp<<<(N_EDGES + 255) / 256, 256, 0, stream>>>(dst, alpha, amax, denom);

  // 6: weighted scatter-aggregate (dominant, L2-resident)
  gat_aggregate<<<N_EDGES, 128, 0, stream>>>(src, dst, alpha, denom, h, out_agg);

  // 7: output projection + tanh
  gat_out_proj<<<(N_NODES * 32 + 255) / 256, 256, 0, stream>>>(out_agg, W_out, y);
}